// MultiBoxLoss_1151051235602
// MI455X (gfx1250) — compile-verified
//
#include <hip/hip_runtime.h>
#include <hip/hip_bf16.h>
#include <math.h>

#define NB      8
#define NP      19248
#define NOBJ    8
#define MD      32
#define NC      81
#define PHW     138
#define PPIX    (PHW*PHW)     // 19044
#define SHW     69
#define IMSZ    550
#define KSEL    100
#define KPAD    112           // 7 tiles of 16
#define NTILES  7
#define MTILES  1191          // ceil(19044/16)
#define GEMM_BX 149           // ceil(1191/8 waves)
#define BBOX_BLOCKS 1024
#define SEM_BLOCKS  19

typedef float v2f __attribute__((ext_vector_type(2)));
typedef float v8f __attribute__((ext_vector_type(8)));

// ---------------------------------------------------------------- K1: matching
__global__ __launch_bounds__(256) void k_match(
    const float* __restrict__ targets, const float* __restrict__ priors,
    float* __restrict__ bto, int* __restrict__ bti, int* __restrict__ conf_t,
    int* __restrict__ npos)
{
  const int b = blockIdx.x;
  const int tid = threadIdx.x;
  __shared__ float sT[NOBJ*5];
  if (tid < NOBJ*5) sT[tid] = targets[b*NOBJ*5 + tid];
  __syncthreads();

  float* btoB = bto + b*NP;
  int*   btiB = bti + b*NP;

  // phase 1: best truth per prior (ties -> lowest truth index)
  for (int p = tid; p < NP; p += 256) {
    float pcx = priors[p*4+0], pcy = priors[p*4+1];
    float pw  = priors[p*4+2], ph  = priors[p*4+3];
    float px1 = pcx - pw*0.5f, py1 = pcy - ph*0.5f;
    float px2 = pcx + pw*0.5f, py2 = pcy + ph*0.5f;
    float pa = (px2-px1)*(py2-py1);
    float best = -1e30f; int bj = 0;
    for (int j = 0; j < NOBJ; ++j) {
      float t0=sT[j*5],t1=sT[j*5+1],t2=sT[j*5+2],t3=sT[j*5+3];
      float iw = fmaxf(fminf(t2,px2) - fmaxf(t0,px1), 0.f);
      float ih = fmaxf(fminf(t3,py2) - fmaxf(t1,py1), 0.f);
      float inter = iw*ih;
      float ta = (t2-t0)*(t3-t1);
      float iou = inter / (ta + pa - inter);
      if (iou > best) { best = iou; bj = j; }
    }
    btoB[p] = best; btiB[p] = bj;
  }
  __syncthreads();

  // phase 2: greedy bipartite matching (replicates the lax.scan semantics)
  __shared__ int   remCol[NOBJ];
  __shared__ int   remRow[NOBJ];
  __shared__ int   nRem;
  __shared__ float rv[256];
  __shared__ int   ri[256];
  __shared__ float bpo[NOBJ];
  __shared__ int   bpi[NOBJ];
  if (tid < NOBJ) remRow[tid] = 0;
  if (tid == 0) nRem = 0;
  __syncthreads();

  for (int it = 0; it < NOBJ; ++it) {
    for (int j = 0; j < NOBJ; ++j) {
      float bv = -1e30f; int bidx = 0;
      if (!remRow[j]) {
        float t0=sT[j*5],t1=sT[j*5+1],t2=sT[j*5+2],t3=sT[j*5+3];
        float ta=(t2-t0)*(t3-t1);
        for (int p = tid; p < NP; p += 256) {
          bool skip = false;
          for (int r = 0; r < nRem; ++r) if (remCol[r] == p) skip = true;
          if (skip) continue;
          float pcx=priors[p*4],pcy=priors[p*4+1],pw=priors[p*4+2],ph=priors[p*4+3];
          float px1=pcx-pw*0.5f,py1=pcy-ph*0.5f,px2=pcx+pw*0.5f,py2=pcy+ph*0.5f;
          float pa=(px2-px1)*(py2-py1);
          float iw=fmaxf(fminf(t2,px2)-fmaxf(t0,px1),0.f);
          float ih=fmaxf(fminf(t3,py2)-fmaxf(t1,py1),0.f);
          float inter=iw*ih;
          float iou = inter/(ta+pa-inter);
          if (iou > bv) { bv = iou; bidx = p; }   // ascending p -> first max kept
        }
      }
      rv[tid]=bv; ri[tid]=bidx;
      __syncthreads();
      for (int off=128; off>0; off>>=1) {
        if (tid < off) {
          float v2=rv[tid+off]; int i2=ri[tid+off];
          if (v2 > rv[tid] || (v2 == rv[tid] && i2 < ri[tid])) { rv[tid]=v2; ri[tid]=i2; }
        }
        __syncthreads();
      }
      if (tid == 0) { bpo[j] = remRow[j] ? -1.0f : rv[0]; bpi[j] = ri[0]; }
      __syncthreads();
    }
    if (tid == 0) {
      float mv = -1e30f; int mj = 0;
      for (int j = 0; j < NOBJ; ++j) if (bpo[j] > mv) { mv = bpo[j]; mj = j; }
      int mi = bpi[mj];
      remCol[nRem] = mi; nRem = nRem + 1;
      remRow[mj] = 1;
      btoB[mi] = 2.0f; btiB[mi] = mj;
    }
    __syncthreads();
  }

  // phase 3: conf labels + positive count
  int cnt = 0;
  for (int p = tid; p < NP; p += 256) {
    float ov = btoB[p]; int j = btiB[p];
    int c = (int)sT[j*5+4] + 1;
    if (ov < 0.5f) c = -1;
    if (ov < 0.4f) c = 0;
    conf_t[b*NP + p] = c;
    if (c > 0) cnt++;
  }
  __shared__ int sc[256];
  sc[tid] = cnt; __syncthreads();
  for (int off=128; off>0; off>>=1) { if (tid<off) sc[tid]+=sc[tid+off]; __syncthreads(); }
  if (tid == 0) npos[b] = sc[0];
}

// ---------------------------------------------------------------- K2: bbox smooth-L1
__global__ __launch_bounds__(256) void k_bbox(
    const float* __restrict__ loc, const float* __restrict__ targets,
    const float* __restrict__ priors, const int* __restrict__ bti,
    const int* __restrict__ conf_t, float* __restrict__ partial)
{
  int tid = threadIdx.x;
  float s = 0.f;
  for (int i = blockIdx.x*256 + tid; i < NB*NP; i += BBOX_BLOCKS*256) {
    if (conf_t[i] > 0) {
      int b = i / NP, p = i % NP;
      int j = bti[i];
      const float* t = targets + (b*NOBJ + j)*5;
      float pcx=priors[p*4],pcy=priors[p*4+1],pw=priors[p*4+2],ph=priors[p*4+3];
      float g[4];
      g[0] = ((t[0]+t[2])*0.5f - pcx) / (0.1f*pw);
      g[1] = ((t[1]+t[3])*0.5f - pcy) / (0.1f*ph);
      g[2] = logf((t[2]-t[0]) / pw) / 0.2f;
      g[3] = logf((t[3]-t[1]) / ph) / 0.2f;
      for (int c = 0; c < 4; ++c) {
        float d = loc[i*4+c] - g[c];
        float ad = fabsf(d);
        s += (ad < 1.f) ? 0.5f*d*d : ad - 0.5f;
      }
    }
  }
  __shared__ float sR[256];
  sR[tid]=s; __syncthreads();
  for (int off=128; off>0; off>>=1){ if(tid<off) sR[tid]+=sR[tid+off]; __syncthreads(); }
  if (tid==0) partial[blockIdx.x] = sR[0];
}

// ---------------------------------------------------------------- K3: top-100 positive selection
__global__ void k_select(
    const float* __restrict__ targets, const int* __restrict__ bti,
    const int* __restrict__ conf_t,
    int* __restrict__ selPrior, int* __restrict__ selObj, int* __restrict__ selValid,
    float* __restrict__ selX1, float* __restrict__ selX2,
    float* __restrict__ selY1, float* __restrict__ selY2,
    float* __restrict__ selScale)
{
  int b = blockIdx.x;
  if (threadIdx.x != 0) return;
  int cnt = 0;
  int tmp[KSEL];
  for (int p = 0; p < NP && cnt < KSEL; ++p)
    if (conf_t[b*NP + p] > 0) tmp[cnt++] = p;
  for (int k = 0; k < KPAD; ++k) {
    int o = b*KPAD + k;
    if (k < cnt) {
      int p = tmp[k];
      int j = bti[b*NP + p];
      const float* t = targets + (b*NOBJ + j)*5;
      float b0=t[0],b1=t[1],b2=t[2],b3=t[3];
      selPrior[o]=p; selObj[o]=j; selValid[o]=1;
      selX1[o]=fmaxf(fminf(b0,b2)*(float)PHW - 1.f, 0.f);
      selX2[o]=fminf(fmaxf(b0,b2)*(float)PHW + 1.f, (float)PHW);
      selY1[o]=fmaxf(fminf(b1,b3)*(float)PHW - 1.f, 0.f);
      selY2[o]=fminf(fmaxf(b1,b3)*(float)PHW + 1.f, (float)PHW);
      float gw = fmaxf((b2-b0)*(float)PHW, 1e-4f);
      float gh = fmaxf((b3-b1)*(float)PHW, 1e-4f);
      selScale[o] = (float)PPIX/(gw*gh);
    } else {
      selPrior[o]=0; selObj[o]=0; selValid[o]=0;
      selX1[o]=0.f; selX2[o]=0.f; selY1[o]=0.f; selY2[o]=0.f; selScale[o]=0.f;
    }
  }
}

// ---------------------------------------------------------------- K4: separable antialiased resize profiles
__device__ float resProf(int o, int outSize, float lo, float hi)
{
  float inv = (float)IMSZ / (float)outSize;       // kernel_scale = 1/scale
  float c = ((float)o + 0.5f)*inv - 0.5f;         // sample_f
  int ilo = (int)floorf(c - inv); if (ilo < 0) ilo = 0;
  int ihi = (int)ceilf(c + inv);  if (ihi > IMSZ-1) ihi = IMSZ-1;
  float wsum = 0.f, win = 0.f;
  for (int i = ilo; i <= ihi; ++i) {
    float wgt = 1.0f - fabsf(c - (float)i)/inv;   // triangle kernel
    if (wgt > 0.f) {
      wsum += wgt;
      float g = ((float)i + 0.5f)/(float)IMSZ;
      if (g >= lo && g < hi) win += wgt;
    }
  }
  return win / wsum;
}

__global__ __launch_bounds__(160) void k_prof(
    const float* __restrict__ targets,
    float* __restrict__ px138, float* __restrict__ py138,
    float* __restrict__ px69,  float* __restrict__ py69)
{
  int b = blockIdx.x / NOBJ, n = blockIdx.x % NOBJ;
  const float* t = targets + (b*NOBJ + n)*5;
  int o = threadIdx.x;
  if (o < PHW) {
    px138[(b*NOBJ+n)*PHW + o] = resProf(o, PHW, t[0], t[2]);
    py138[(b*NOBJ+n)*PHW + o] = resProf(o, PHW, t[1], t[3]);
  }
  if (o < SHW) {
    px69[(b*NOBJ+n)*SHW + o] = resProf(o, SHW, t[0], t[2]);
    py69[(b*NOBJ+n)*SHW + o] = resProf(o, SHW, t[1], t[3]);
  }
}

// ---------------------------------------------------------------- K5: mask GEMM (WMMA f32) + BCE
__global__ __launch_bounds__(256) void k_mask(
    const float* __restrict__ proto, const float* __restrict__ mask_data,
    const int* __restrict__ selPrior, const int* __restrict__ selObj,
    const int* __restrict__ selValid,
    const float* __restrict__ selX1, const float* __restrict__ selX2,
    const float* __restrict__ selY1, const float* __restrict__ selY2,
    const float* __restrict__ px138, const float* __restrict__ py138,
    float* __restrict__ maskPartial)
{
  const int b  = blockIdx.z;
  const int nt = blockIdx.y;
  const int tid = threadIdx.x;
  const int lane = tid & 31;
  const int wave = tid >> 5;
  const int l16 = lane & 15;
  const int hi  = lane >> 4;

  __shared__ float sCoef[16][MD];
  __shared__ float sPX[NOBJ][PHW];
  __shared__ float sPY[NOBJ][PHW];
  __shared__ float sX1[16], sX2[16], sY1[16], sY2[16];
  __shared__ int   sObj[16], sVal[16];
  __shared__ float sWS[8][16];

  for (int e = tid; e < 16*MD; e += 256) {
    int n = e >> 5, d = e & 31;
    int o = b*KPAD + nt*16 + n;
    float v = 0.f;
    if (selValid[o]) v = mask_data[(b*NP + selPrior[o])*MD + d];
    sCoef[n][d] = v;
  }
  for (int e = tid; e < NOBJ*PHW; e += 256) {
    int n = e / PHW, x = e % PHW;
    sPX[n][x] = px138[(b*NOBJ+n)*PHW + x];
    sPY[n][x] = py138[(b*NOBJ+n)*PHW + x];
  }
  if (tid < 16) {
    int o = b*KPAD + nt*16 + tid;
    sX1[tid]=selX1[o]; sX2[tid]=selX2[o]; sY1[tid]=selY1[o]; sY2[tid]=selY2[o];
    sObj[tid]=selObj[o]; sVal[tid]=selValid[o];
  }
  __syncthreads();

  const int mtile = blockIdx.x*8 + wave;
  const int m0 = mtile*16;
  int mrow = m0 + l16;
  int mclamp = mrow < PPIX ? mrow : (PPIX-1);
  const float* arow = proto + ((size_t)b*PPIX + (size_t)mclamp)*MD;

  v8f acc = {0.f,0.f,0.f,0.f,0.f,0.f,0.f,0.f};
#pragma unroll
  for (int s = 0; s < 8; ++s) {
    v2f a, bb;
    a.x  = arow[4*s + 2*hi];
    a.y  = arow[4*s + 2*hi + 1];
    bb.x = sCoef[l16][4*s + 2*hi];
    bb.y = sCoef[l16][4*s + 2*hi + 1];
    acc = __builtin_amdgcn_wmma_f32_16x16x4_f32(
        /*neg_a=*/false, a, /*neg_b=*/false, bb,
        /*c_mod=*/(short)0, acc, /*reuse_a=*/false, /*reuse_b=*/false);
  }

  // BCE over this wave's 16 pixels for mask N = l16 (C rows: M = v + 8*hi)
  float sum = 0.f;
  const int n = l16;
  const bool nval = sVal[n] != 0;
  const float x1 = sX1[n], x2 = sX2[n], y1 = sY1[n], y2 = sY2[n];
  const int obj = sObj[n];
  if (nval) {
#pragma unroll
    for (int v = 0; v < 8; ++v) {
      int pix = m0 + v + 8*hi;
      if (pix < PPIX) {
        int y = pix / PHW, x = pix - y*PHW;
        float z = acc[v];
        float pr = 1.0f / (1.0f + expf(-z));
        float fx = (float)x, fy = (float)y;
        bool keep = (fx >= x1) && (fx < x2) && (fy >= y1) && (fy < y2);
        float pe = keep ? pr : 0.0f;
        pe = fminf(fmaxf(pe, 1e-7f), 1.0f - 1e-7f);
        float tgt = (sPY[obj][y]*sPX[obj][x] > 0.5f) ? 1.0f : 0.0f;
        sum += -(tgt*logf(pe) + (1.0f - tgt)*logf(1.0f - pe));
      }
    }
  }
  sum += __shfl_xor(sum, 16, 32);
  if (lane < 16) sWS[wave][lane] = sum;
  __syncthreads();
  if (tid < 16) {
    float t = 0.f;
    for (int w = 0; w < 8; ++w) t += sWS[w][tid];
    maskPartial[(size_t)((b*NTILES + nt)*16 + tid)*GEMM_BX + blockIdx.x] = t;
  }
}

// ---------------------------------------------------------------- K6: logsumexp + masked loss_c keys
__global__ __launch_bounds__(256) void k_lse(
    const float* __restrict__ conf_data, const int* __restrict__ conf_t,
    float* __restrict__ lse, float* __restrict__ key)
{
  int tid = threadIdx.x, lane = tid & 31, wave = tid >> 5;
  for (int row = blockIdx.x*8 + wave; row < NB*NP; row += 2048*8) {
    const float* lg = conf_data + (size_t)row*NC;
    float m = -INFINITY;
    for (int c = lane; c < NC; c += 32) m = fmaxf(m, lg[c]);
    for (int off = 16; off > 0; off >>= 1) m = fmaxf(m, __shfl_xor(m, off, 32));
    float s = 0.f;
    for (int c = lane; c < NC; c += 32) s += expf(lg[c] - m);
    for (int off = 16; off > 0; off >>= 1) s += __shfl_xor(s, off, 32);
    float L = m + logf(s);
    if (lane == 0) {
      lse[row] = L;
      int ct = conf_t[row];
      key[row] = (ct != 0) ? 0.f : (L - lg[0]);   // zero at pos / neutral
    }
  }
}

// ---------------------------------------------------------------- K7: hard-negative mining + CE (per batch)
__global__ __launch_bounds__(1024) void k_neg(
    const float* __restrict__ conf_data, const int* __restrict__ conf_t,
    const float* __restrict__ lse, const float* __restrict__ key,
    const int* __restrict__ npos, float* __restrict__ confLoss)
{
  const int b = blockIdx.x;
  const int tid = threadIdx.x;
  __shared__ int   sCount;
  __shared__ int   sScan[1024];
  __shared__ float sRed[1024];
  __shared__ int   sRun;
  const float* keyB = key + b*NP;

  int np = npos[b];
  int K = np*3; if (K > NP-1) K = NP-1;

  // radix select: T = K-th largest key (IEEE bits monotone, key >= 0)
  unsigned T = 0u;
  for (int bit = 31; bit >= 0; --bit) {
    unsigned cand = T | (1u << bit);
    if (tid == 0) sCount = 0;
    __syncthreads();
    int c = 0;
    for (int p = tid; p < NP; p += 1024)
      if (__float_as_uint(keyB[p]) >= cand) c++;
    atomicAdd(&sCount, c);
    __syncthreads();
    if (sCount >= K) T = cand;     // uniform decision (shared value)
    __syncthreads();
  }
  if (tid == 0) sCount = 0;
  __syncthreads();
  int cg = 0;
  for (int p = tid; p < NP; p += 1024)
    if (__float_as_uint(keyB[p]) > T) cg++;
  atomicAdd(&sCount, cg);
  __syncthreads();
  const int quota = K - sCount;     // equals (value==T) selected by lowest index
  __syncthreads();
  if (tid == 0) sRun = 0;
  __syncthreads();

  float ce = 0.f;
  for (int base = 0; base < NP; base += 1024) {
    int p = base + tid;
    unsigned u = (p < NP) ? __float_as_uint(keyB[p]) : 0u;
    int eq = (p < NP && u == T) ? 1 : 0;
    sScan[tid] = eq; __syncthreads();
    for (int off = 1; off < 1024; off <<= 1) {
      int v = (tid >= off) ? sScan[tid - off] : 0;
      __syncthreads();
      sScan[tid] += v;
      __syncthreads();
    }
    int excl = sScan[tid] - eq;
    int tot  = sScan[1023];
    int run  = sRun;
    bool selEq = eq && (run + excl < quota);
    if (p < NP) {
      int ct = conf_t[b*NP + p];
      bool pos = ct > 0;
      bool neg = ((u > T) || selEq) && !pos && (ct >= 0);
      if (pos || neg) {
        int cls = pos ? ct : 0;
        ce += lse[b*NP + p] - conf_data[(size_t)(b*NP + p)*NC + cls];
      }
    }
    __syncthreads();
    if (tid == 0) sRun = run + tot;
    __syncthreads();
  }
  sRed[tid] = ce; __syncthreads();
  for (int off = 512; off > 0; off >>= 1) {
    if (tid < off) sRed[tid] += sRed[tid + off];
    __syncthreads();
  }
  if (tid == 0) confLoss[b] = sRed[0];
}

// ---------------------------------------------------------------- K8: semantic segmentation loss
__global__ __launch_bounds__(256) void k_sem(
    const float* __restrict__ segm, const float* __restrict__ targets,
    const float* __restrict__ px69, const float* __restrict__ py69,
    float* __restrict__ partial)
{
  const int b = blockIdx.z;
  const int tid = threadIdx.x;
  __shared__ int sLab[NOBJ];
  if (tid < NOBJ) sLab[tid] = (int)targets[(b*NOBJ + tid)*5 + 4];
  __syncthreads();
  float sum = 0.f;
  for (int pix = blockIdx.x*256 + tid; pix < SHW*SHW; pix += SEM_BLOCKS*256) {
    int y = pix / SHW, x = pix % SHW;
    bool t[NOBJ];
    for (int n = 0; n < NOBJ; ++n)
      t[n] = py69[(b*NOBJ+n)*SHW + y] * px69[(b*NOBJ+n)*SHW + x] > 0.5f;
    for (int c = 0; c < NC-1; ++c) {
      bool st = false;
      for (int n = 0; n < NOBJ; ++n) if (sLab[n] == c && t[n]) st = true;
      float s = segm[((size_t)b*(NC-1) + c)*SHW*SHW + pix];
      sum += fmaxf(s, 0.f) - (st ? s : 0.f) + log1pf(expf(-fabsf(s)));
    }
  }
  __shared__ float sR[256];
  sR[tid] = sum; __syncthreads();
  for (int off = 128; off > 0; off >>= 1) { if (tid < off) sR[tid] += sR[tid+off]; __syncthreads(); }
  if (tid == 0) partial[b*SEM_BLOCKS + blockIdx.x] = sR[0];
}

// ---------------------------------------------------------------- K9: final deterministic combine
__global__ __launch_bounds__(1024) void k_final(
    const int* __restrict__ npos, const float* __restrict__ bboxPartial,
    const float* __restrict__ confLoss, const float* __restrict__ semPartial,
    const float* __restrict__ maskPartial, const float* __restrict__ selScale,
    float* __restrict__ out)
{
  const int tid = threadIdx.x;
  __shared__ float sR[1024];
  sR[tid] = bboxPartial[tid];
  __syncthreads();
  for (int off = 512; off > 0; off >>= 1) { if (tid < off) sR[tid] += sR[tid+off]; __syncthreads(); }
  float lossB = sR[0];
  __syncthreads();

  float mv = 0.f;
  if (tid < NB*KPAD) {
    int b = tid / KPAD, n = tid % KPAD;
    int nt = n >> 4, nl = n & 15;
    const float* mp = maskPartial + (size_t)((b*NTILES + nt)*16 + nl)*GEMM_BX;
    float s = 0.f;
    for (int k = 0; k < GEMM_BX; ++k) s += mp[k];
    mv = s * selScale[b*KPAD + n];
  }
  sR[tid] = mv; __syncthreads();
  for (int off = 512; off > 0; off >>= 1) { if (tid < off) sR[tid] += sR[tid+off]; __syncthreads(); }

  if (tid == 0) {
    float lossM = sR[0];
    float tp = 0.f;
    for (int b = 0; b < NB; ++b) tp += (float)npos[b];
    tp = fmaxf(tp, 1.0f);
    float lossC = 0.f;
    for (int b = 0; b < NB; ++b) lossC += confLoss[b];
    float lossS = 0.f;
    for (int i = 0; i < NB*SEM_BLOCKS; ++i) lossS += semPartial[i];
    out[0] = lossB * 1.5f / tp;
    out[1] = lossC / tp;
    out[2] = lossM * 6.125f / (float)PPIX / tp;
    out[3] = lossS / ((float)SHW*(float)SHW) / (float)NB;
  }
}

// ---------------------------------------------------------------- launch
extern "C" void kernel_launch(void* const* d_in, const int* in_sizes, int n_in,
                              void* d_out, int out_size, void* d_ws, size_t ws_size,
                              hipStream_t stream)
{
  (void)in_sizes; (void)n_in; (void)out_size; (void)ws_size;
  const float* loc    = (const float*)d_in[0];
  const float* conf   = (const float*)d_in[1];
  const float* maskd  = (const float*)d_in[2];
  const float* proto  = (const float*)d_in[3];
  const float* segm   = (const float*)d_in[4];
  const float* priors = (const float*)d_in[5];
  const float* targ   = (const float*)d_in[6];
  // d_in[7] = gt_masks: reconstructed analytically from targets (box masks are
  // separable, so the antialiased resize factorizes into 1-D profiles).

  char* w = (char*)d_ws;
  auto alloc = [&](size_t bytes) -> void* {
    void* p = (void*)w;
    w += (bytes + 255) & ~(size_t)255;
    return p;
  };
  float* bto       = (float*)alloc((size_t)NB*NP*4);
  int*   bti       = (int*)  alloc((size_t)NB*NP*4);
  int*   conf_t    = (int*)  alloc((size_t)NB*NP*4);
  float* lse       = (float*)alloc((size_t)NB*NP*4);
  float* key       = (float*)alloc((size_t)NB*NP*4);
  int*   npos      = (int*)  alloc(NB*4);
  int*   selPrior  = (int*)  alloc(NB*KPAD*4);
  int*   selObj    = (int*)  alloc(NB*KPAD*4);
  int*   selValid  = (int*)  alloc(NB*KPAD*4);
  float* selX1     = (float*)alloc(NB*KPAD*4);
  float* selX2     = (float*)alloc(NB*KPAD*4);
  float* selY1     = (float*)alloc(NB*KPAD*4);
  float* selY2     = (float*)alloc(NB*KPAD*4);
  float* selScale  = (float*)alloc(NB*KPAD*4);
  float* px138     = (float*)alloc((size_t)NB*NOBJ*PHW*4);
  float* py138     = (float*)alloc((size_t)NB*NOBJ*PHW*4);
  float* px69      = (float*)alloc((size_t)NB*NOBJ*SHW*4);
  float* py69      = (float*)alloc((size_t)NB*NOBJ*SHW*4);
  float* maskPart  = (float*)alloc((size_t)NB*NTILES*16*GEMM_BX*4);
  float* bboxPart  = (float*)alloc(BBOX_BLOCKS*4);
  float* confLoss  = (float*)alloc(NB*4);
  float* semPart   = (float*)alloc(NB*SEM_BLOCKS*4);

  k_match <<<NB, 256, 0, stream>>>(targ, priors, bto, bti, conf_t, npos);
  k_bbox  <<<BBOX_BLOCKS, 256, 0, stream>>>(loc, targ, priors, bti, conf_t, bboxPart);
  k_select<<<NB, 32, 0, stream>>>(targ, bti, conf_t, selPrior, selObj, selValid,
                                  selX1, selX2, selY1, selY2, selScale);
  k_prof  <<<NB*NOBJ, 160, 0, stream>>>(targ, px138, py138, px69, py69);
  dim3 g5(GEMM_BX, NTILES, NB);
  k_mask  <<<g5, 256, 0, stream>>>(proto, maskd, selPrior, selObj, selValid,
                                   selX1, selX2, selY1, selY2, px138, py138, maskPart);
  k_lse   <<<2048, 256, 0, stream>>>(conf, conf_t, lse, key);
  k_neg   <<<NB, 1024, 0, stream>>>(conf, conf_t, lse, key, npos, confLoss);
  dim3 g8(SEM_BLOCKS, 1, NB);
  k_sem   <<<g8, 256, 0, stream>>>(segm, targ, px69, py69, semPart);
  k_final <<<1, 1024, 0, stream>>>(npos, bboxPart, confLoss, semPart, maskPart,
                                   selScale, (float*)d_out);
}